// SAGE_OneHot_MLP_42150809043597
// MI455X (gfx1250) — compile-verified
//
#include <hip/hip_runtime.h>

typedef float v2f __attribute__((ext_vector_type(2)));
typedef float v8f __attribute__((ext_vector_type(8)));

// ---------------------------------------------------------------- utilities

__global__ void zero_f32(float* __restrict__ p, long n) {
  long i = (long)blockIdx.x * blockDim.x + threadIdx.x;
  long stride = (long)gridDim.x * blockDim.x;
  for (; i < n; i += stride) p[i] = 0.0f;
}

__global__ void count_edges(const long long* __restrict__ tgt,
                            float* __restrict__ cnt, int E) {
  int i = blockIdx.x * blockDim.x + threadIdx.x;
  int stride = gridDim.x * blockDim.x;
  for (; i < E; i += stride) atomicAdd(&cnt[(int)tgt[i]], 1.0f);
}

__global__ void finalize_inv(float* __restrict__ cnt, int n) {
  int i = blockIdx.x * blockDim.x + threadIdx.x;
  if (i < n) cnt[i] = 1.0f / fmaxf(cnt[i], 1.0f);
}

// one thread per (edge, feature); consecutive threads -> consecutive features
__global__ void scatter_add(const long long* __restrict__ src,
                            const long long* __restrict__ tgt,
                            const float* __restrict__ feat,
                            float* __restrict__ agg, int E, int logd) {
  long total = (long)E << logd;
  long i = (long)blockIdx.x * blockDim.x + threadIdx.x;
  long stride = (long)gridDim.x * blockDim.x;
  int dmask = (1 << logd) - 1;
  for (; i < total; i += stride) {
    long e = i >> logd;
    int f = (int)(i & dmask);
    int s = (int)src[e];
    int t = (int)tgt[e];
    atomicAdd(&agg[((long)t << logd) + f], feat[((long)s << logd) + f]);
  }
}

// ---------------------------------------------------------------- WMMA GEMM
// Out[N,M] = maybe_relu( (rowscale.*A1) @ W1^T  [+ A2 @ W2^T]  + bias )
// A[N,K] row-major, W[M,K] row-major (so B = W^T is [K,M]).
// One wave32 computes one 16x16 output tile with V_WMMA_F32_16X16X4_F32.
// 16x4 f32 A fragment (ISA 7.12.2): lanes 0-15 hold K=k0,k0+1 ; lanes 16-31
// hold K=k0+2,k0+3 for row M=lane%16. B mirrored with N in place of M.
// C/D: VGPR j holds row M=j (lanes 0-15) and M=j+8 (lanes 16-31), N=lane%16.
__global__ void __launch_bounds__(128)
wmma_gemm(const float* __restrict__ A1, const float* __restrict__ rowscale,
          const float* __restrict__ W1,
          const float* __restrict__ A2, const float* __restrict__ W2,
          const float* __restrict__ bias, float* __restrict__ Out,
          int N, int K, int M, int relu) {
  int mt = M >> 4;
  int tiles = (N >> 4) * mt;
  int tile = blockIdx.x * 4 + threadIdx.y;   // wave-uniform
  if (tile >= tiles) return;                 // whole wave exits -> EXEC stays all-1s
  int tm = tile / mt;
  int tn = tile - tm * mt;

  int lane = threadIdx.x;
  int half = lane >> 4;
  int l16  = lane & 15;
  int arow = (tm << 4) + l16;                // row fed into A fragment
  int bcol = (tn << 4) + l16;                // output column / W row fed into B

  const float* a1p = A1 + (long)arow * K + 2 * half;
  const float* w1p = W1 + (long)bcol * K + 2 * half;
  float s = rowscale ? rowscale[arow] : 1.0f;

  v8f c = {0.f, 0.f, 0.f, 0.f, 0.f, 0.f, 0.f, 0.f};
  for (int k0 = 0; k0 < K; k0 += 4) {
    v2f a, b;
    a[0] = a1p[k0] * s;
    a[1] = a1p[k0 + 1] * s;
    b[0] = w1p[k0];
    b[1] = w1p[k0 + 1];
    c = __builtin_amdgcn_wmma_f32_16x16x4_f32(false, a, false, b,
                                              (short)0, c, false, false);
  }
  if (A2) {
    const float* a2p = A2 + (long)arow * K + 2 * half;
    const float* w2p = W2 + (long)bcol * K + 2 * half;
    for (int k0 = 0; k0 < K; k0 += 4) {
      v2f a, b;
      a[0] = a2p[k0];
      a[1] = a2p[k0 + 1];
      b[0] = w2p[k0];
      b[1] = w2p[k0 + 1];
      c = __builtin_amdgcn_wmma_f32_16x16x4_f32(false, a, false, b,
                                                (short)0, c, false, false);
    }
  }

  float bv = bias ? bias[bcol] : 0.0f;
#pragma unroll
  for (int j = 0; j < 8; ++j) {
    int orow = (tm << 4) + j + 8 * half;
    float v = c[j] + bv;
    if (relu) v = fmaxf(v, 0.0f);
    Out[(long)orow * M + bcol] = v;
  }
}

// ---------------------------------------------------------------- batchnorm

// grid.x == C, block == 256 -> 256 accumulating threads per column
__global__ void col_stats(const float* __restrict__ H, float* __restrict__ sum,
                          float* __restrict__ sumsq, int N, int C) {
  int tid = blockIdx.x * blockDim.x + threadIdx.x;
  int T = gridDim.x * blockDim.x;
  int c = tid & (C - 1);                 // C is a power of two (64/128)
  int r0 = tid / C;
  int rstride = T / C;
  float s = 0.f, q = 0.f;
  for (int r = r0; r < N; r += rstride) {
    float v = H[(long)r * C + c];
    s += v;
    q += v * v;
  }
  atomicAdd(&sum[c], s);
  atomicAdd(&sumsq[c], q);
}

__global__ void bn_apply_relu(float* __restrict__ H, const float* __restrict__ sum,
                              const float* __restrict__ sumsq,
                              const float* __restrict__ gamma,
                              const float* __restrict__ beta, int N, int C) {
  long total = (long)N * C;
  long i = (long)blockIdx.x * blockDim.x + threadIdx.x;
  long stride = (long)gridDim.x * blockDim.x;
  float invN = 1.0f / (float)N;
  for (; i < total; i += stride) {
    int c = (int)(i & (C - 1));
    float mu = sum[c] * invN;
    float var = fmaxf(sumsq[c] * invN - mu * mu, 0.0f);
    float v = (H[i] - mu) * rsqrtf(var + 1e-5f) * gamma[c] + beta[c];
    H[i] = fmaxf(v, 0.0f);
  }
}

__global__ void fc3_kernel(const float* __restrict__ Z, const float* __restrict__ w,
                           const float* __restrict__ b, float* __restrict__ out,
                           int N, int C) {
  int i = blockIdx.x * blockDim.x + threadIdx.x;
  if (i >= N) return;
  float acc = b[0];
  for (int j = 0; j < C; ++j) acc += Z[(long)i * C + j] * w[j];
  out[i] = acc;
}

// ---------------------------------------------------------------- launcher

extern "C" void kernel_launch(void* const* d_in, const int* in_sizes, int n_in,
                              void* d_out, int out_size, void* d_ws, size_t ws_size,
                              hipStream_t stream) {
  const float*     x     = (const float*)d_in[0];
  const long long* ei    = (const long long*)d_in[1];
  const float*     w_l1  = (const float*)d_in[2];
  const float*     b_l1  = (const float*)d_in[3];
  const float*     w_r1  = (const float*)d_in[4];
  const float*     w_l2  = (const float*)d_in[5];
  const float*     b_l2  = (const float*)d_in[6];
  const float*     w_r2  = (const float*)d_in[7];
  const float*     fc1_w = (const float*)d_in[8];
  const float*     fc1_b = (const float*)d_in[9];
  const float*     bn1_g = (const float*)d_in[10];
  const float*     bn1_b = (const float*)d_in[11];
  const float*     fc2_w = (const float*)d_in[12];
  const float*     fc2_b = (const float*)d_in[13];
  const float*     bn2_g = (const float*)d_in[14];
  const float*     bn2_b = (const float*)d_in[15];
  const float*     fc3_w = (const float*)d_in[16];
  const float*     fc3_b = (const float*)d_in[17];

  const int N = 100000;
  const int E = in_sizes[1] / 2;
  const long long* src = ei;
  const long long* tgt = ei + E;

  // d_out = [out (N) | h1 (N*64) | h2 (N*64)]
  float* outv = (float*)d_out;
  float* h1   = outv + N;
  float* h2   = h1 + (long)N * 64;

  // workspace
  float* ws   = (float*)d_ws;
  float* inv  = ws;                          // N     (degree -> inverse)
  float* agg  = inv + N;                     // N*64  (aggregation; reused for fc2)
  float* bufA = agg + (long)N * 64;          // N*128 (fc1 pre-BN / post-BN)
  float* st   = bufA + (long)N * 128;        // 512 stats floats
  float* sum1 = st;
  float* sq1  = st + 128;
  float* sum2 = st + 256;
  float* sq2  = st + 320;

  dim3 b256(256);
  auto gsz = [](long n, int bs) { return dim3((unsigned)((n + bs - 1) / bs)); };

  // degree counts -> inverse
  zero_f32<<<gsz(N, 256), b256, 0, stream>>>(inv, N);
  count_edges<<<gsz(E, 256), b256, 0, stream>>>(tgt, inv, E);
  finalize_inv<<<gsz(N, 256), b256, 0, stream>>>(inv, N);

  // ---- SAGEConv 1: mean-agg of x (d=32), then fused dual GEMM + bias + relu
  zero_f32<<<gsz((long)N * 32, 256), b256, 0, stream>>>(agg, (long)N * 32);
  scatter_add<<<gsz((long)E * 32, 256), b256, 0, stream>>>(src, tgt, x, agg, E, 5);
  {
    int tiles = (N / 16) * (64 / 16);
    wmma_gemm<<<dim3((tiles + 3) / 4), dim3(32, 4), 0, stream>>>(
        agg, inv, w_l1, x, w_r1, b_l1, h1, N, 32, 64, 1);
  }

  // ---- SAGEConv 2: mean-agg of h1 (d=64)
  zero_f32<<<gsz((long)N * 64, 256), b256, 0, stream>>>(agg, (long)N * 64);
  scatter_add<<<gsz((long)E * 64, 256), b256, 0, stream>>>(src, tgt, h1, agg, E, 6);
  {
    int tiles = (N / 16) * (64 / 16);
    wmma_gemm<<<dim3((tiles + 3) / 4), dim3(32, 4), 0, stream>>>(
        agg, inv, w_l2, h1, w_r2, b_l2, h2, N, 64, 64, 1);
  }

  // ---- fc1 (pre-BN) -> bufA
  {
    int tiles = (N / 16) * (128 / 16);
    wmma_gemm<<<dim3((tiles + 3) / 4), dim3(32, 4), 0, stream>>>(
        h2, nullptr, fc1_w, nullptr, nullptr, fc1_b, bufA, N, 64, 128, 0);
  }
  zero_f32<<<gsz(512, 256), b256, 0, stream>>>(st, 512);
  col_stats<<<dim3(128), b256, 0, stream>>>(bufA, sum1, sq1, N, 128);
  bn_apply_relu<<<gsz((long)N * 128, 256), b256, 0, stream>>>(
      bufA, sum1, sq1, bn1_g, bn1_b, N, 128);

  // ---- fc2 (pre-BN) -> agg (reused)
  {
    int tiles = (N / 16) * (64 / 16);
    wmma_gemm<<<dim3((tiles + 3) / 4), dim3(32, 4), 0, stream>>>(
        bufA, nullptr, fc2_w, nullptr, nullptr, fc2_b, agg, N, 128, 64, 0);
  }
  col_stats<<<dim3(64), b256, 0, stream>>>(agg, sum2, sq2, N, 64);
  bn_apply_relu<<<gsz((long)N * 64, 256), b256, 0, stream>>>(
      agg, sum2, sq2, bn2_g, bn2_b, N, 64);

  // ---- fc3 -> out
  fc3_kernel<<<gsz(N, 256), b256, 0, stream>>>(agg, fc3_w, fc3_b, outv, N, 64);
}